// ChamferLoss_18537078849647
// MI455X (gfx1250) — compile-verified
//
#include <hip/hip_runtime.h>

typedef float v2f __attribute__((ext_vector_type(2)));
typedef float v8f __attribute__((ext_vector_type(8)));

#define NPTS  8192
#define NBATCH 4
#define TILES (NPTS / 16)

// One wave owns 16 "query" points and sweeps all 8192 "ref" points in 16-wide
// tiles. Each tile is ONE v_wmma_f32_16x16x4_f32:
//   A row m = (-2*rx, -2*ry, -2*rz, |r|^2)   (ref tile, reloaded per iter)
//   B col n = ( qx,   qy,   qz,   1   )      (query tile, loop-invariant)
//   C       = |q|^2 broadcast per column
//   D = A*B + C = |r|^2 + |q|^2 - 2 r.q  ==  squared distance tile.
// D layout: VGPR v, lane l -> ref row (v + 8*(l>=16)), query col (l&15).
__global__ __launch_bounds__(256)
void chamfer_wmma_kernel(const float* __restrict__ preds,
                         const float* __restrict__ gts,
                         float* __restrict__ out)
{
    const int lane = threadIdx.x & 31;
    const int wave = threadIdx.x >> 5;
    const int b    = blockIdx.y;      // batch
    const int dir  = blockIdx.z;      // 0: queries=preds refs=gts ; 1: swapped

    const float* __restrict__ q = (dir ? gts   : preds) + (size_t)b * NPTS * 3;
    const float* __restrict__ r = (dir ? preds : gts)   + (size_t)b * NPTS * 3;

    const int hi = lane >> 4;         // which K-half / ref-half this lane holds
    const int m  = lane & 15;         // row/col index within 16-tile
    const int qi = blockIdx.x * 128 + wave * 16 + m;   // global query index

    // ---- loop-invariant B (query tile) and C (|q|^2 broadcast) ----
    const float qx = q[qi * 3 + 0];
    const float qy = q[qi * 3 + 1];
    const float qz = q[qi * 3 + 2];
    const float rq = qx * qx + qy * qy + qz * qz;

    v2f B;
    B.x = hi ? qz   : qx;   // K=2 : K=0
    B.y = hi ? 1.0f : qy;   // K=3 : K=1  (pad slot multiplies A's |r|^2)

    v8f C;
#pragma unroll
    for (int v = 0; v < 8; ++v) C[v] = rq;

    float best[8];
    int   btile[8];                   // winning TILE number per slot (scalar src)
#pragma unroll
    for (int v = 0; v < 8; ++v) { best[v] = __builtin_inff(); btile[v] = 0; }

    // ---- sweep all ref tiles (branchless body: EXEC stays all-ones) ----
#pragma unroll 4
    for (int t = 0; t < TILES; ++t) {
        const int rb = t * 16;
        const float* rp = r + (size_t)(rb + m) * 3;
        const float cx = rp[0], cy = rp[1], cz = rp[2];

        // prefetch ~8 tiles ahead (speculative: dropped if invalid)
        __builtin_prefetch(r + (size_t)(rb + 128 + m) * 3, 0, 3);

        const float nx = -2.0f * cx;
        const float ny = -2.0f * cy;
        const float nz = -2.0f * cz;
        const float nr = cx * cx + cy * cy + cz * cz;

        v2f A;                        // cndmask, no divergent branch
        A.x = hi ? nz : nx;           // K=2 : K=0
        A.y = hi ? nr : ny;           // K=3 : K=1

        v8f P = __builtin_amdgcn_wmma_f32_16x16x4_f32(
            /*neg_a=*/false, A, /*neg_b=*/false, B,
            /*c_mod=*/(short)0, C, /*reuse_a=*/false, /*reuse_b=*/false);

#pragma unroll
        for (int v = 0; v < 8; ++v) {
            const float p = P[v];
            const bool  c = p < best[v];     // strict: first occurrence wins
            best[v]  = c ? p : best[v];
            btile[v] = c ? t : btile[v];     // track tile only (scalar operand)
        }
    }

    // ---- reconstruct full ref indices, then reduce the 8 per-lane slots ----
    int bidx[8];
#pragma unroll
    for (int v = 0; v < 8; ++v) bidx[v] = btile[v] * 16 + v + 8 * hi;

    float bv = best[0];
    int   bi = bidx[0];
#pragma unroll
    for (int v = 1; v < 8; ++v) {
        const bool c = best[v] < bv;   // ascending index order => '<' keeps first
        bv = c ? best[v] : bv;
        bi = c ? bidx[v] : bi;
    }

    // ---- combine the two ref-halves held by lanes (l, l+16) ----
    const float ov = __shfl_xor(bv, 16, 32);
    const int   oi = __shfl_xor(bi, 16, 32);
    const bool take = (ov < bv) || (ov == bv && oi < bi);  // tie -> lower index
    bv = take ? ov : bv;
    bi = take ? oi : bi;

    if (!hi) {
        const int S = NBATCH * NPTS;               // 32768 elements per section
        const int o = b * NPTS + qi;
        out[(dir ? S : 0) + o] = bv;               // min sections 0,1
        int* iout = (int*)out;
        iout[2 * S + (dir ? S : 0) + o] = bi;      // argmin sections 2,3 (int32)
    }
}

extern "C" void kernel_launch(void* const* d_in, const int* in_sizes, int n_in,
                              void* d_out, int out_size, void* d_ws, size_t ws_size,
                              hipStream_t stream)
{
    const float* preds = (const float*)d_in[0];   // [4, 8192, 3]
    const float* gts   = (const float*)d_in[1];   // [4, 8192, 3]
    // d_in[2] = mask, unused by the reference computation.
    float* out = (float*)d_out;                   // 4 * 32768 elements

    dim3 grid(NPTS / 128, NBATCH, 2);             // 128 queries per 256-thr block
    dim3 block(256);
    chamfer_wmma_kernel<<<grid, block, 0, stream>>>(preds, gts, out);
}